// RGCLayer_77249281786066
// MI455X (gfx1250) — compile-verified
//
#include <hip/hip_runtime.h>
#include <hip/hip_bf16.h>
#include <stdint.h>

// ---------------------------------------------------------------------------
// CDNA5 (gfx1250) helpers
// ---------------------------------------------------------------------------

// Low 32 bits of a generic pointer to a __shared__ object are the LDS byte
// offset (generic->local addrspacecast on AMDGPU is a truncate).
__device__ __forceinline__ unsigned lds_off(const void* p) {
    return (unsigned)(uintptr_t)p;
}

// GLOBAL_LOAD_ASYNC_TO_LDS_B32: per-lane async DMA global -> LDS (ASYNCcnt).
__device__ __forceinline__ void async_copy_b32(unsigned lds_dst, const void* src) {
    asm volatile("global_load_async_to_lds_b32 %0, %1, off"
                 :: "v"(lds_dst), "v"(src)
                 : "memory");
}

__device__ __forceinline__ void wait_async0() {
    asm volatile("s_wait_asynccnt 0" ::: "memory");
}

// Hardware f32 atomic add, no return (STOREcnt), saddr form:
//   global_atomic_add_f32 v_offset, v_data, s[base:base+1] [offset:4]
// Scalar base per edge (SALU), constant per-lane byte offset (lane*8).
__device__ __forceinline__ void gatomic_fadd_s0(float* base, unsigned voff, float v) {
    asm volatile("global_atomic_add_f32 %0, %1, %2"
                 :: "v"(voff), "v"(v), "s"(base)
                 : "memory");
}
__device__ __forceinline__ void gatomic_fadd_s4(float* base, unsigned voff, float v) {
    asm volatile("global_atomic_add_f32 %0, %1, %2 offset:4"
                 :: "v"(voff), "v"(v), "s"(base)
                 : "memory");
}

// ---------------------------------------------------------------------------
// Fused SpMM: one pass over an edge list does both
//     orow[r] += v * xcol[c]      (A   @ xcol)
//     ocol[c] += v * xrow[r]      (A^T @ xrow)
// Edge triples are streamed into LDS with double-buffered async copies.
// Block = 256 threads = 8 waves; chunk = 256 edges; wave w consumes exactly
// the 32 edges its own lanes async-loaded, so per-wave s_wait_asynccnt 0 is
// the only synchronization needed — zero workgroup barriers.
// Wave-per-edge: lane l covers embed dims 2l, 2l+1 (wave32 * 2 = 64).
// r/c are wave-uniform -> readfirstlane moves all address math to SALU and
// both gathers and atomics use scalar-base addressing.
// ---------------------------------------------------------------------------
#define CHUNK 256

__global__ void __launch_bounds__(256)
spmm_fused_kernel(const int*   __restrict__ rows,
                  const int*   __restrict__ cols,
                  const float* __restrict__ vals,
                  int nnz,
                  const float* __restrict__ xcol,   // gathered by col index
                  const float* __restrict__ xrow,   // gathered by row index
                  float*       __restrict__ orow,   // scattered by row index
                  float*       __restrict__ ocol)   // scattered by col index
{
    __shared__ int   lrows[2][CHUNK];
    __shared__ int   lcols[2][CHUNK];
    __shared__ float lvals[2][CHUNK];

    const int tid  = threadIdx.x;
    const int lane = tid & 31;
    const int wv   = tid >> 5;

    const unsigned voff  = (unsigned)lane << 3;   // lane * 8 bytes (float2)
    const int      lelem = lane << 1;             // lane * 2 floats

    const int nchunks = (nnz + CHUNK - 1) / CHUNK;

    // Issue this thread's 3 async B32 copies for one chunk into buffer `buf`.
    auto issue = [&](int chunk, int buf) {
        int e = chunk * CHUNK + tid;
        if (e < nnz) {
            async_copy_b32(lds_off(&lrows[buf][tid]), rows + e);
            async_copy_b32(lds_off(&lcols[buf][tid]), cols + e);
            async_copy_b32(lds_off(&lvals[buf][tid]), vals + e);
        }
    };

    int chunk = blockIdx.x;
    if (chunk < nchunks) issue(chunk, 0);

    int buf = 0;
    for (; chunk < nchunks; chunk += gridDim.x) {
        wait_async0();                      // this wave's current buffer is in LDS

        int next = chunk + gridDim.x;
        if (next < nchunks) issue(next, buf ^ 1);   // prefetch depth-1

        const int base = chunk * CHUNK;
        int cnt = nnz - base;
        if (cnt > CHUNK) cnt = CHUNK;

        int jend = wv * 32 + 32;
        if (jend > cnt) jend = cnt;

        for (int j = wv * 32; j < jend; ++j) {
            // Uniform LDS address across the wave -> broadcast reads;
            // readfirstlane makes r/c provably uniform (SGPRs).
            const int   r = __builtin_amdgcn_readfirstlane(lrows[buf][j]);
            const int   c = __builtin_amdgcn_readfirstlane(lcols[buf][j]);
            const float v = lvals[buf][j];

            const float* xpb = xcol + ((size_t)c << 6);   // x[col] row base
            const float* xsb = xrow + ((size_t)r << 6);   // x[row] row base

            const float2 xp = *(const float2*)(xpb + lelem);
            const float2 xs = *(const float2*)(xsb + lelem);

            float* pr = orow + ((size_t)r << 6);
            float* pc = ocol + ((size_t)c << 6);

            gatomic_fadd_s0(pr, voff, v * xp.x);
            gatomic_fadd_s4(pr, voff, v * xp.y);
            gatomic_fadd_s0(pc, voff, v * xs.x);
            gatomic_fadd_s4(pc, voff, v * xs.y);
        }
        buf ^= 1;
    }
}

// ---------------------------------------------------------------------------
// Zero-fill the (harness-poisoned) output buffer.
// ---------------------------------------------------------------------------
__global__ void __launch_bounds__(256)
zero_kernel(float4* __restrict__ out, int n4) {
    int i = blockIdx.x * blockDim.x + threadIdx.x;
    if (i < n4) {
        out[i] = make_float4(0.f, 0.f, 0.f, 0.f);
    }
}

// ---------------------------------------------------------------------------
// Launch
// ---------------------------------------------------------------------------
extern "C" void kernel_launch(void* const* d_in, const int* in_sizes, int n_in,
                              void* d_out, int out_size, void* d_ws, size_t ws_size,
                              hipStream_t stream) {
    const float* student = (const float*)d_in[0];   // [NS, 64]
    const float* problem = (const float*)d_in[1];   // [NP, 64]
    const int*   a_rows  = (const int*)  d_in[2];
    const int*   a_cols  = (const int*)  d_in[3];
    const float* a_vals  = (const float*)d_in[4];
    const int*   ia_rows = (const int*)  d_in[5];
    const int*   ia_cols = (const int*)  d_in[6];
    const float* ia_vals = (const float*)d_in[7];

    const int ns64 = in_sizes[0];          // NS * 64
    const int np64 = in_sizes[1];          // NP * 64
    const int nnz  = in_sizes[2];

    float* out = (float*)d_out;
    // Output order: student_c, student_ic, problem_c, problem_ic
    float* student_c  = out;
    float* student_ic = out + (size_t)ns64;
    float* problem_c  = out + (size_t)2 * ns64;
    float* problem_ic = out + (size_t)2 * ns64 + np64;

    // 1) zero-fill outputs (harness poisons d_out)
    {
        int n4 = (out_size + 3) / 4;       // out_size is a multiple of 4 here
        int grid = (n4 + 255) / 256;
        zero_kernel<<<grid, 256, 0, stream>>>((float4*)d_out, n4);
    }

    // 2) fused SpMM passes
    int nchunks = (nnz + CHUNK - 1) / CHUNK;
    int grid = nchunks < 2048 ? nchunks : 2048;
    if (grid < 1) grid = 1;

    // edge set a: student_c = A @ P ; problem_c = A^T @ S
    spmm_fused_kernel<<<grid, 256, 0, stream>>>(
        a_rows, a_cols, a_vals, nnz, problem, student, student_c, problem_c);

    // edge set ia: student_ic = IA @ P ; problem_ic = IA^T @ S
    spmm_fused_kernel<<<grid, 256, 0, stream>>>(
        ia_rows, ia_cols, ia_vals, nnz, problem, student, student_ic, problem_ic);
}